// LightGCN_78254304133889
// MI455X (gfx1250) — compile-verified
//
#include <hip/hip_runtime.h>

// ---------------------------------------------------------------------------
// LightGCN propagation on gfx1250 (MI455X).
//   out = (h0 + A h0 + A^2 h0 + A^3 h0) / 4,  A given as COO (rows, cols, vals)
// Working set (~155 MB) fits in the 192 MB L2 -> bound by L2 atomic/scatter
// throughput. Scatter kernel: 64 f32 atomics/edge (the floor), edge stream
// staged into LDS with gfx1250 async global->LDS loads (ASYNCcnt-tracked),
// double buffered, retired with s_wait_asynccnt. All indexing is 32-bit;
// batch partitioning (a division) is precomputed on the host.
// ---------------------------------------------------------------------------

#define DIM 64            // embedding dim
#define TB 256            // threads per block (8 wave32)
#define BATCH 1024        // edges staged per block iteration
#define EPW (BATCH / 8)   // edges processed per wave per batch (8 waves/block)

// --- gfx1250 async global->LDS b128 (ASYNCcnt-tracked, no VGPR landing) ----
__device__ __forceinline__ void async_g2l_b128(unsigned int lds_byte_off,
                                               const void* gptr) {
    asm volatile("global_load_async_to_lds_b128 %0, %1, off"
                 :: "v"(lds_byte_off),
                    "v"((unsigned long long)(uintptr_t)gptr)
                 : "memory");
}
// Low 32 bits of a generic pointer to __shared__ == LDS byte offset (ISA §10.2)
__device__ __forceinline__ unsigned int lds_off(const void* p) {
    return (unsigned int)(uintptr_t)p;
}

// COO SpMM scatter: hout[rows[e], :] += vals[e] * hin[cols[e], :]
// - edge stream (rows/cols/vals) staged into LDS with async b128 loads,
//   double buffered; every thread issues exactly 3 asyncs per batch so
//   s_wait_asynccnt 3 retires batch k while batch k+1 is in flight
//   (per-wave async loads complete in order).
// - one wave32 per edge: r/c scalarized via readfirstlane (saddr-form VMEM),
//   each lane owns 2 dims -> 1 global_load_b64 gather + 2 f32 atomics.
__global__ void lgcn_spmm_scatter(const int* __restrict__ rows,
                                  const int* __restrict__ cols,
                                  const float* __restrict__ vals,
                                  const float2* __restrict__ hin2,  // [N*32]
                                  float* __restrict__ hout,         // [N*64]
                                  int nnz, int totalBatches, int bpb) {
    __shared__ int   sRows[2][BATCH];
    __shared__ int   sCols[2][BATCH];
    __shared__ float sVals[2][BATCH];

    const int tid  = threadIdx.x;
    const int lane = tid & 31;
    const int wid  = tid >> 5;                       // 0..7

    const int b0   = (int)blockIdx.x * bpb;
    const int bEnd = (b0 + bpb < totalBatches) ? (b0 + bpb) : totalBatches;
    if (b0 >= bEnd) return;

    // clamp keeps tail-batch async addresses in-bounds & 16B-aligned;
    // garbage slots are never consumed (guarded by emax below)
    const int clampMax = (nnz >= 4) ? ((nnz - 4) & ~3) : 0;

    auto issue = [&](int p, int batch) {
        int i4 = batch * BATCH + tid * 4;
        if (i4 > clampMax) i4 = clampMax;
        async_g2l_b128(lds_off(&sRows[p][tid * 4]), rows + i4);
        async_g2l_b128(lds_off(&sCols[p][tid * 4]), cols + i4);
        async_g2l_b128(lds_off(&sVals[p][tid * 4]), vals + i4);
    };

    issue(0, b0);
    for (int b = b0; b < bEnd; ++b) {
        const int p = (b - b0) & 1;
        const bool hasNext = (b + 1 < bEnd);
        if (hasNext) {
            issue(p ^ 1, b + 1);
            asm volatile("s_wait_asynccnt 0x3" ::: "memory"); // batch b landed
        } else {
            asm volatile("s_wait_asynccnt 0x0" ::: "memory");
        }
        __syncthreads();                                      // visible to all waves

        const int base = b * BATCH;
        const int rem  = nnz - base;
        const int emax = (rem < BATCH) ? rem : BATCH;
        const int lo = wid * EPW;
        const int hi = (lo + EPW < emax) ? (lo + EPW) : emax;

        #pragma unroll 8
        for (int e = lo; e < hi; ++e) {
            int   r = __builtin_amdgcn_readfirstlane(sRows[p][e]);
            int   c = __builtin_amdgcn_readfirstlane(sCols[p][e]);
            float v = sVals[p][e];
            float2 x = hin2[c * 32 + lane];
            float* dst = hout + r * DIM + lane * 2;
            unsafeAtomicAdd(dst + 0, v * x.x);
            unsafeAtomicAdd(dst + 1, v * x.y);
        }
        __syncthreads();       // all reads of buffer p done before its reuse
    }
}

// init: h0 = concat(user, item); acc = h0; h1 = 0
__global__ void lgcn_init(const float4* __restrict__ user4,
                          const float4* __restrict__ item4,
                          float4* __restrict__ h0,
                          float4* __restrict__ acc,
                          float4* __restrict__ h1,
                          int n4u, int n4) {
    int i = blockIdx.x * blockDim.x + threadIdx.x;
    if (i >= n4) return;
    float4 v = (i < n4u) ? user4[i] : item4[i - n4u];
    h0[i]  = v;
    acc[i] = v;
    h1[i]  = make_float4(0.f, 0.f, 0.f, 0.f);
}

// acc += hsrc; hzero = 0   (prepares the other ping-pong buffer)
__global__ void lgcn_accum_zero(float4* __restrict__ acc,
                                const float4* __restrict__ hsrc,
                                float4* __restrict__ hzero,
                                int n4) {
    int i = blockIdx.x * blockDim.x + threadIdx.x;
    if (i >= n4) return;
    float4 a = acc[i];
    float4 s = hsrc[i];
    a.x += s.x; a.y += s.y; a.z += s.z; a.w += s.w;
    acc[i] = a;
    hzero[i] = make_float4(0.f, 0.f, 0.f, 0.f);
}

// acc = (acc + hsrc) * scale    (final layer + mean)
__global__ void lgcn_accum_scale(float4* __restrict__ acc,
                                 const float4* __restrict__ hsrc,
                                 int n4, float scale) {
    int i = blockIdx.x * blockDim.x + threadIdx.x;
    if (i >= n4) return;
    float4 a = acc[i];
    float4 s = hsrc[i];
    a.x = (a.x + s.x) * scale;
    a.y = (a.y + s.y) * scale;
    a.z = (a.z + s.z) * scale;
    a.w = (a.w + s.w) * scale;
    acc[i] = a;
}

extern "C" void kernel_launch(void* const* d_in, const int* in_sizes, int n_in,
                              void* d_out, int out_size, void* d_ws, size_t ws_size,
                              hipStream_t stream) {
    const float* user_emb = (const float*)d_in[0];   // [U, 64]
    const float* item_emb = (const float*)d_in[1];   // [I, 64]
    const int*   rows     = (const int*)d_in[2];     // [NNZ]
    const int*   cols     = (const int*)d_in[3];     // [NNZ]
    const float* vals     = (const float*)d_in[4];   // [NNZ]

    const int Un  = in_sizes[0] / DIM;
    const int In  = in_sizes[1] / DIM;
    const int Nn  = Un + In;
    const int nnz = in_sizes[2];

    float* acc = (float*)d_out;                      // [N, 64] accumulator == output
    float* h0  = (float*)d_ws;                       // [N, 64]
    float* h1  = h0 + (size_t)Nn * DIM;              // [N, 64]

    const int n4  = Nn * (DIM / 4);                  // float4 elements per buffer
    const int n4u = Un * (DIM / 4);
    const int eb  = (n4 + TB - 1) / TB;              // elementwise grid
    const int sb  = 1024;                            // scatter grid

    // batch partitioning precomputed on host (no device integer division)
    const int totalBatches = (nnz + BATCH - 1) / BATCH;
    const int bpb = (totalBatches + sb - 1) / sb;

    // h0 = all_emb, acc = all_emb, h1 = 0
    lgcn_init<<<eb, TB, 0, stream>>>((const float4*)user_emb, (const float4*)item_emb,
                                     (float4*)h0, (float4*)acc, (float4*)h1, n4u, n4);

    // layer 1: h1 = A h0 ; acc += h1 ; h0 = 0
    lgcn_spmm_scatter<<<sb, TB, 0, stream>>>(rows, cols, vals, (const float2*)h0, h1,
                                             nnz, totalBatches, bpb);
    lgcn_accum_zero<<<eb, TB, 0, stream>>>((float4*)acc, (const float4*)h1, (float4*)h0, n4);

    // layer 2: h0 = A h1 ; acc += h0 ; h1 = 0
    lgcn_spmm_scatter<<<sb, TB, 0, stream>>>(rows, cols, vals, (const float2*)h1, h0,
                                             nnz, totalBatches, bpb);
    lgcn_accum_zero<<<eb, TB, 0, stream>>>((float4*)acc, (const float4*)h0, (float4*)h1, n4);

    // layer 3: h1 = A h0 ; acc = (acc + h1) / 4
    lgcn_spmm_scatter<<<sb, TB, 0, stream>>>(rows, cols, vals, (const float2*)h0, h1,
                                             nnz, totalBatches, bpb);
    lgcn_accum_scale<<<eb, TB, 0, stream>>>((float4*)acc, (const float4*)h1, n4, 0.25f);
}